// ImprovedMultiHeadDiscriminator_44470091383439
// MI455X (gfx1250) — compile-verified
//
#include <hip/hip_runtime.h>
#include <hip/hip_bf16.h>
#include <stdint.h>

// ---------------------------------------------------------------------------
// GAT x3 + attention pooling + MLP heads for MI455X (gfx1250, wave32, WMMA).
//
// Input flattening assumption (setup_inputs() dict insertion order, recursive):
//   0 node_features[50000,128] f32    1 edge_index[2,300000] i32
//   2 edge_attr[300000,128] f32       3 batch[50000] i32
//   4 core_path_mask[50000] bool(1B)  5 question_emb[64,128] f32
//   6..11  gat0 {W,We,a_src,a_dst,a_edge,b}
//   12..17 gat1, 18..23 gat2
//   24..27 path_att {W1,b1,W2,b2}
//   28..33 structural {W1,b1,W2,b2,W3,b3}, 34..39 semantic, 40..45 logic
//   46..49 qenc {W1,b1,W2,b2}
// Output: [4,64,1] f32 -> d_out[i*64+g].
// ---------------------------------------------------------------------------

#define NNODES 50000
#define NEDGES 300000
#define NGRAPH 64
#define EMB    128

typedef float v2f __attribute__((ext_vector_type(2)));
typedef float v8f __attribute__((ext_vector_type(8)));

__device__ __forceinline__ float lrelu02(float x) { return x > 0.f ? x : 0.2f * x; }

// Order-preserving float<->uint key for atomicMax-based segment max.
// Key 0 == "less than anything" sentinel (so memset(0) initializes).
__device__ __forceinline__ unsigned fkey(float f) {
    unsigned b = __float_as_uint(f);
    return (b & 0x80000000u) ? ~b : (b | 0x80000000u);
}
__device__ __forceinline__ float funkey(unsigned k) {
    unsigned b = (k & 0x80000000u) ? (k & 0x7FFFFFFFu) : ~k;
    return __uint_as_float(b);
}

__device__ __forceinline__ float wave_sum32(float v) {
#pragma unroll
    for (int o = 16; o > 0; o >>= 1) v += __shfl_xor(v, o, 32);
    return v;
}

// ---------------------------------------------------------------------------
// WMMA fp32 GEMM: C[M,N] = A[M,K] * W[K,N].  M%16==0, K%4==0, N%64==0.
// Block = 8 waves (32,8); each wave computes a 16x64 strip using
// V_WMMA_F32_16X16X4_F32 fragments (ISA 7.12.2 layouts).
// ---------------------------------------------------------------------------
__global__ __launch_bounds__(256) void gemm_wmma_f32(
    const float* __restrict__ A, const float* __restrict__ W,
    float* __restrict__ C, int M, int K, int N)
{
    const int lane  = threadIdx.x;
    const int mtile = blockIdx.x * 8 + threadIdx.y;
    if (mtile * 16 >= M) return;
    const int row0 = mtile * 16;
    const int n0   = blockIdx.y * 64;
    const int half = lane >> 4;   // 0: lanes 0-15, 1: lanes 16-31
    const int l16  = lane & 15;

    v8f acc[4] = {};
    const float* arow = A + (size_t)(row0 + l16) * K + 2 * half;

    for (int k0 = 0; k0 < K; k0 += 4) {
        // A 16x4 f32 frag: lanes0-15 hold K={k0,k0+1}, lanes16-31 K={k0+2,k0+3}
        v2f a = *reinterpret_cast<const v2f*>(arow + k0);
        // B 4x16 f32 frag: VGPR0 = row (k0+2*half), VGPR1 = row (k0+2*half+1)
        const float* wr0 = W + (size_t)(k0 + 2 * half) * N + n0 + l16;
        const float* wr1 = wr0 + N;
#pragma unroll
        for (int j = 0; j < 4; ++j) {
            v2f b;
            b[0] = wr0[j * 16];
            b[1] = wr1[j * 16];
            acc[j] = __builtin_amdgcn_wmma_f32_16x16x4_f32(
                false, a, false, b, (short)0, acc[j], false, false);
        }
    }
#pragma unroll
    for (int j = 0; j < 4; ++j) {
        const int col = n0 + j * 16 + l16;
#pragma unroll
        for (int r = 0; r < 8; ++r) {
            const int row = row0 + r + 8 * half;  // C: VGPR r -> M=r / M=8+r
            C[(size_t)row * N + col] = acc[j][r];
        }
    }
}

// ---------------------------------------------------------------------------
// Self-loop attr = mean of incoming edge attrs
// ---------------------------------------------------------------------------
__global__ void k_deg(const int* __restrict__ dst, float* __restrict__ deg, int E) {
    int e = blockIdx.x * blockDim.x + threadIdx.x;
    if (e < E) atomicAdd(&deg[dst[e]], 1.0f);
}
__global__ void k_loop_acc(const float* __restrict__ ea, const int* __restrict__ dst,
                           float* __restrict__ la, int E) {
    long idx = (long)blockIdx.x * blockDim.x + threadIdx.x;
    if (idx >= (long)E * EMB) return;
    int e = (int)(idx >> 7), c = (int)(idx & 127);
    atomicAdd(&la[(size_t)dst[e] * EMB + c], ea[idx]);
}
__global__ void k_loop_div(float* __restrict__ la, const float* __restrict__ deg, int N) {
    long idx = (long)blockIdx.x * blockDim.x + threadIdx.x;
    if (idx >= (long)N * EMB) return;
    la[idx] /= fmaxf(deg[idx >> 7], 1.0f);
}

// vfold[fi*4+h] = sum_c We[fi, h*128+c] * a_edge[h,c]  (fold of edge projection)
__global__ void k_vfold(const float* __restrict__ We, const float* __restrict__ aeg,
                        float* __restrict__ vf, int H, int HC) {
    int t = blockIdx.x * blockDim.x + threadIdx.x;
    if (t >= EMB * 4) return;
    int fi = t >> 2, h = t & 3;
    float s = 0.f;
    if (h < H) {
        const float* w = We + (size_t)fi * HC + h * 128;
        const float* a = aeg + h * 128;
        for (int c = 0; c < 128; ++c) s += w[c] * a[c];
    }
    vf[t] = s;
}

// s_src/s_dst[n,h] = xs[n,h,:] . a_src/a_dst[h,:]   (wave per node)
__global__ __launch_bounds__(256) void k_sdots(
    const float* __restrict__ xs, const float* __restrict__ asr,
    const float* __restrict__ ads, float* __restrict__ ssrc,
    float* __restrict__ sdst, int N, int H, int HC)
{
    int n = blockIdx.x * 8 + threadIdx.y;
    if (n >= N) return;
    int lane = threadIdx.x;
    const float4* xr = reinterpret_cast<const float4*>(xs + (size_t)n * HC);
    for (int h = 0; h < H; ++h) {
        float4 xv = xr[h * 32 + lane];
        float4 as = reinterpret_cast<const float4*>(asr + h * 128)[lane];
        float4 ad = reinterpret_cast<const float4*>(ads + h * 128)[lane];
        float ps = xv.x * as.x + xv.y * as.y + xv.z * as.z + xv.w * as.w;
        float pd = xv.x * ad.x + xv.y * ad.y + xv.z * ad.z + xv.w * ad.w;
        ps = wave_sum32(ps);
        pd = wave_sum32(pd);
        if (lane == 0) { ssrc[n * H + h] = ps; sdst[n * H + h] = pd; }
    }
}

// sc[r,h] = attr[r,:] . vfold[:,h]   (wave per row; used for edges and loops)
__global__ __launch_bounds__(256) void k_escore(
    const float* __restrict__ attr, const float* __restrict__ vf,
    float* __restrict__ sc, int rows, int H)
{
    int r = blockIdx.x * 8 + threadIdx.y;
    if (r >= rows) return;
    int lane = threadIdx.x;
    float4 av = reinterpret_cast<const float4*>(attr + (size_t)r * EMB)[lane];
    int c0 = lane * 4;
    for (int h = 0; h < H; ++h) {
        float p = av.x * vf[(c0 + 0) * 4 + h] + av.y * vf[(c0 + 1) * 4 + h] +
                  av.z * vf[(c0 + 2) * 4 + h] + av.w * vf[(c0 + 3) * 4 + h];
        p = wave_sum32(p);
        if (lane == 0) sc[r * H + h] = p;
    }
}

// attention: leaky-relu score + segment max (edges, then self-loops)
__global__ void k_attn_max_e(const int* __restrict__ src, const int* __restrict__ dst,
                             const float* __restrict__ ssrc, const float* __restrict__ sdst,
                             float* __restrict__ esc, unsigned* __restrict__ amax,
                             int E, int H) {
    int idx = blockIdx.x * blockDim.x + threadIdx.x;
    if (idx >= E * H) return;
    int e = idx / H, h = idx - e * H;
    int s = src[e], d = dst[e];
    float a = lrelu02(ssrc[s * H + h] + sdst[d * H + h] + esc[idx]);
    esc[idx] = a;
    atomicMax(&amax[d * H + h], fkey(a));
}
__global__ void k_attn_max_l(const float* __restrict__ ssrc, const float* __restrict__ sdst,
                             float* __restrict__ lsc, unsigned* __restrict__ amax,
                             int N, int H) {
    int idx = blockIdx.x * blockDim.x + threadIdx.x;
    if (idx >= N * H) return;
    float a = lrelu02(ssrc[idx] + sdst[idx] + lsc[idx]);
    lsc[idx] = a;
    atomicMax(&amax[idx], fkey(a));
}
__global__ void k_attn_sum_e(const int* __restrict__ dst, float* __restrict__ esc,
                             const unsigned* __restrict__ amax, float* __restrict__ denom,
                             int E, int H) {
    int idx = blockIdx.x * blockDim.x + threadIdx.x;
    if (idx >= E * H) return;
    int e = idx / H, h = idx - e * H;
    int d = dst[e];
    float w = __expf(esc[idx] - funkey(amax[d * H + h]));
    esc[idx] = w;
    atomicAdd(&denom[d * H + h], w);
}
__global__ void k_attn_sum_l(float* __restrict__ lsc, const unsigned* __restrict__ amax,
                             float* __restrict__ denom, int N, int H) {
    int idx = blockIdx.x * blockDim.x + threadIdx.x;
    if (idx >= N * H) return;
    float w = __expf(lsc[idx] - funkey(amax[idx]));
    lsc[idx] = w;
    atomicAdd(&denom[idx], w);
}

// out[dst] += (w/denom) * xs[src]   (block per edge)
__global__ __launch_bounds__(256) void k_agg_e(
    const int* __restrict__ src, const int* __restrict__ dst,
    const float* __restrict__ esc, const float* __restrict__ denom,
    const float* __restrict__ xs, float* __restrict__ out, int H, int HC)
{
    __shared__ float coeff[4];
    int e = blockIdx.x, tid = threadIdx.x;
    int s = src[e], d = dst[e];
    if (tid < H) coeff[tid] = esc[e * H + tid] / denom[d * H + tid];
    __syncthreads();
    const float* xr = xs + (size_t)s * HC;
    float* orow = out + (size_t)d * HC;
    for (int j = tid; j < HC; j += 256) atomicAdd(&orow[j], coeff[j >> 7] * xr[j]);
}
// self loops: out[n] += coeff * xs[n] (no atomics needed, separate launch)
__global__ void k_agg_l(const float* __restrict__ lsc, const float* __restrict__ denom,
                        const float* __restrict__ xs, float* __restrict__ out,
                        int N, int H, int hcShift)
{
    long idx = (long)blockIdx.x * blockDim.x + threadIdx.x;
    if (idx >= ((long)N << hcShift)) return;
    int n = (int)(idx >> hcShift);
    int h = (int)((idx & ((1 << hcShift) - 1)) >> 7);
    out[idx] += (lsc[n * H + h] / denom[n * H + h]) * xs[idx];
}

__global__ void k_activate(const float* __restrict__ acc, const float* __restrict__ b,
                           float* __restrict__ x, int N, int hcShift) {
    long idx = (long)blockIdx.x * blockDim.x + threadIdx.x;
    if (idx >= ((long)N << hcShift)) return;
    x[idx] = fmaxf(acc[idx] + b[(int)(idx & ((1 << hcShift) - 1))], 0.f);
}

// ---------------------------------------------------------------------------
// Pooling: node score MLP, per-graph softmax, weighted graphlet
// ---------------------------------------------------------------------------
__global__ __launch_bounds__(64) void k_pscore(
    const float* __restrict__ x, const float* __restrict__ W1,
    const float* __restrict__ b1, const float* __restrict__ W2,
    const float* __restrict__ b2, const unsigned char* __restrict__ mask,
    float* __restrict__ nscore)
{
    __shared__ float xs_sh[128];
    __shared__ float red[64];
    int n = blockIdx.x, tid = threadIdx.x;
    xs_sh[tid]      = x[(size_t)n * 128 + tid];
    xs_sh[tid + 64] = x[(size_t)n * 128 + 64 + tid];
    __syncthreads();
    float h = b1[tid];
    for (int k = 0; k < 128; ++k) h += xs_sh[k] * W1[k * 64 + tid];
    red[tid] = fmaxf(h, 0.f) * W2[tid];
    __syncthreads();
    for (int s = 32; s > 0; s >>= 1) { if (tid < s) red[tid] += red[tid + s]; __syncthreads(); }
    if (tid == 0) nscore[n] = red[0] + b2[0] + 2.0f * (mask[n] ? 1.0f : 0.0f);
}
__global__ void k_gmax(const float* __restrict__ sc, const int* __restrict__ batch,
                       unsigned* __restrict__ gmax, int N) {
    int n = blockIdx.x * blockDim.x + threadIdx.x;
    if (n < N) atomicMax(&gmax[batch[n]], fkey(sc[n]));
}
__global__ void k_gsum(float* __restrict__ sc, const int* __restrict__ batch,
                       const unsigned* __restrict__ gmax, float* __restrict__ gsum, int N) {
    int n = blockIdx.x * blockDim.x + threadIdx.x;
    if (n >= N) return;
    float w = __expf(sc[n] - funkey(gmax[batch[n]]));
    sc[n] = w;
    atomicAdd(&gsum[batch[n]], w);
}
__global__ void k_graphlet(const float* __restrict__ x, const float* __restrict__ sc,
                           const float* __restrict__ gsum, const int* __restrict__ batch,
                           float* __restrict__ gl, int N) {
    long idx = (long)blockIdx.x * blockDim.x + threadIdx.x;
    if (idx >= (long)N * 128) return;
    int n = (int)(idx >> 7), c = (int)(idx & 127);
    int g = batch[n];
    atomicAdd(&gl[g * 128 + c], (sc[n] / gsum[g]) * x[idx]);
}

// ---------------------------------------------------------------------------
// MLP heads, one block (128 threads) per graph
// ---------------------------------------------------------------------------
__device__ float mlp3_eval(const float* in1, const float* in2, int n2,
                           const float* W1, const float* b1,
                           const float* W2, const float* b2,
                           const float* W3, const float* b3,
                           float* h1, float* red, int tid)
{
    float a = b1[tid];
    for (int k = 0; k < 128; ++k) a += in1[k] * W1[k * 128 + tid];
    if (in2) for (int k = 0; k < n2; ++k) a += in2[k] * W1[(128 + k) * 128 + tid];
    __syncthreads();
    h1[tid] = fmaxf(a, 0.f);
    __syncthreads();
    float r2 = 0.f;
    if (tid < 64) {
        float b = b2[tid];
        for (int k = 0; k < 128; ++k) b += h1[k] * W2[k * 64 + tid];
        r2 = fmaxf(b, 0.f) * W3[tid];
    }
    red[tid] = (tid < 64) ? r2 : 0.f;
    __syncthreads();
    for (int s = 64; s > 0; s >>= 1) { if (tid < s) red[tid] += red[tid + s]; __syncthreads(); }
    float out = red[0] + b3[0];
    __syncthreads();
    return out;
}

__global__ __launch_bounds__(128) void k_heads(
    const float* __restrict__ graphlet, const float* __restrict__ qemb,
    const float* qW1, const float* qb1, const float* qW2, const float* qb2,
    const float* sW1, const float* sb1, const float* sW2, const float* sb2,
    const float* sW3, const float* sb3,
    const float* mW1, const float* mb1, const float* mW2, const float* mb2,
    const float* mW3, const float* mb3,
    const float* lW1, const float* lb1, const float* lW2, const float* lb2,
    const float* lW3, const float* lb3,
    float* __restrict__ out)
{
    __shared__ float gl[128], qe[128], q1[128], q2[128], h1[128], red[128];
    int g = blockIdx.x, tid = threadIdx.x;
    gl[tid] = graphlet[g * 128 + tid];
    qe[tid] = qemb[g * 128 + tid];
    __syncthreads();
    // qenc: q2 = relu(qe@W1+b1)@W2 + b2
    float a = qb1[tid];
    for (int k = 0; k < 128; ++k) a += qe[k] * qW1[k * 128 + tid];
    q1[tid] = fmaxf(a, 0.f);
    __syncthreads();
    a = qb2[tid];
    for (int k = 0; k < 128; ++k) a += q1[k] * qW2[k * 128 + tid];
    q2[tid] = a;
    __syncthreads();

    float s_st = mlp3_eval(gl, nullptr, 0, sW1, sb1, sW2, sb2, sW3, sb3, h1, red, tid);
    float s_lo = mlp3_eval(gl, nullptr, 0, lW1, lb1, lW2, lb2, lW3, lb3, h1, red, tid);
    float s_se = mlp3_eval(gl, q2, 128, mW1, mb1, mW2, mb2, mW3, mb3, h1, red, tid);
    if (tid == 0) {
        out[0 * NGRAPH + g] = s_st;
        out[1 * NGRAPH + g] = s_se;
        out[2 * NGRAPH + g] = s_lo;
        out[3 * NGRAPH + g] = s_st + s_se + s_lo;
    }
}

// ---------------------------------------------------------------------------
extern "C" void kernel_launch(void* const* d_in, const int* in_sizes, int n_in,
                              void* d_out, int out_size, void* d_ws, size_t ws_size,
                              hipStream_t stream) {
    const int N = NNODES, E = NEDGES;
    const float* node_f = (const float*)d_in[0];
    const int*   src    = (const int*)d_in[1];
    const int*   dst    = src + E;
    const float* eattr  = (const float*)d_in[2];
    const int*   batch  = (const int*)d_in[3];
    const unsigned char* mask = (const unsigned char*)d_in[4];  // jax bool = 1 byte
    const float* qemb   = (const float*)d_in[5];
    auto F = [&](int i) { return (const float*)d_in[i]; };

    // workspace bump allocator (256B aligned)
    size_t off = 0;
    auto alloc = [&](size_t bytes) {
        void* p = (char*)d_ws + off;
        off += (bytes + 255) & ~(size_t)255;
        return p;
    };
    float*    B_X   = (float*)alloc((size_t)N * 512 * 4);
    float*    B_XS  = (float*)alloc((size_t)N * 512 * 4);
    float*    B_OUT = (float*)alloc((size_t)N * 512 * 4);
    float*    lattr = (float*)alloc((size_t)N * EMB * 4);
    float*    deg   = (float*)alloc((size_t)N * 4);
    float*    ssrc  = (float*)alloc((size_t)N * 4 * 4);
    float*    sdst  = (float*)alloc((size_t)N * 4 * 4);
    float*    esc   = (float*)alloc((size_t)E * 4 * 4);
    float*    lsc   = (float*)alloc((size_t)N * 4 * 4);
    unsigned* amax  = (unsigned*)alloc((size_t)N * 4 * 4);
    float*    denom = (float*)alloc((size_t)N * 4 * 4);
    float*    vfold = (float*)alloc(EMB * 4 * 4);
    float*    nsc   = (float*)alloc((size_t)N * 4);
    unsigned* gmax  = (unsigned*)alloc(NGRAPH * 4);
    float*    gsum  = (float*)alloc(NGRAPH * 4);
    float*    glbuf = (float*)alloc(NGRAPH * 128 * 4);

    // --- self-loop attrs (mean of incoming edge attrs) ---
    hipMemsetAsync(deg, 0, (size_t)N * 4, stream);
    hipMemsetAsync(lattr, 0, (size_t)N * EMB * 4, stream);
    k_deg<<<(E + 255) / 256, 256, 0, stream>>>(dst, deg, E);
    k_loop_acc<<<(int)(((long)E * EMB + 255) / 256), 256, 0, stream>>>(eattr, dst, lattr, E);
    k_loop_div<<<(int)(((long)N * EMB + 255) / 256), 256, 0, stream>>>(lattr, deg, N);

    // --- 3 GAT layers ---
    const int Fin[3] = {128, 512, 256};
    const int Hh[3]  = {4, 2, 1};
    const int HC[3]  = {512, 256, 128};
    const int hcSh[3] = {9, 8, 7};
    const int iW[3] = {6, 12, 18};

    for (int l = 0; l < 3; ++l) {
        const float* Wl  = F(iW[l] + 0);
        const float* Wel = F(iW[l] + 1);
        const float* asr = F(iW[l] + 2);
        const float* ads = F(iW[l] + 3);
        const float* aeg = F(iW[l] + 4);
        const float* bl  = F(iW[l] + 5);
        const int H = Hh[l], hc = HC[l], K = Fin[l];
        const float* xin = (l == 0) ? node_f : B_X;

        hipMemsetAsync(B_OUT, 0, (size_t)N * hc * 4, stream);
        hipMemsetAsync(denom, 0, (size_t)N * H * 4, stream);
        hipMemsetAsync(amax,  0, (size_t)N * H * 4, stream);

        k_vfold<<<2, 256, 0, stream>>>(Wel, aeg, vfold, H, hc);

        dim3 gblk(32, 8);
        dim3 ggrd((N / 16 + 7) / 8, hc / 64);
        gemm_wmma_f32<<<ggrd, gblk, 0, stream>>>(xin, Wl, B_XS, N, K, hc);

        k_sdots<<<(N + 7) / 8, gblk, 0, stream>>>(B_XS, asr, ads, ssrc, sdst, N, H, hc);
        k_escore<<<(E + 7) / 8, gblk, 0, stream>>>(eattr, vfold, esc, E, H);
        k_escore<<<(N + 7) / 8, gblk, 0, stream>>>(lattr, vfold, lsc, N, H);

        k_attn_max_e<<<(E * H + 255) / 256, 256, 0, stream>>>(src, dst, ssrc, sdst, esc, amax, E, H);
        k_attn_max_l<<<(N * H + 255) / 256, 256, 0, stream>>>(ssrc, sdst, lsc, amax, N, H);
        k_attn_sum_e<<<(E * H + 255) / 256, 256, 0, stream>>>(dst, esc, amax, denom, E, H);
        k_attn_sum_l<<<(N * H + 255) / 256, 256, 0, stream>>>(lsc, amax, denom, N, H);

        k_agg_e<<<E, 256, 0, stream>>>(src, dst, esc, denom, B_XS, B_OUT, H, hc);
        k_agg_l<<<(int)(((long)N * hc + 255) / 256), 256, 0, stream>>>(lsc, denom, B_XS, B_OUT, N, H, hcSh[l]);

        k_activate<<<(int)(((long)N * hc + 255) / 256), 256, 0, stream>>>(B_OUT, bl, B_X, N, hcSh[l]);
    }

    // --- attention pooling over nodes per graph ---
    hipMemsetAsync(gmax, 0, NGRAPH * 4, stream);
    hipMemsetAsync(gsum, 0, NGRAPH * 4, stream);
    hipMemsetAsync(glbuf, 0, NGRAPH * 128 * 4, stream);
    k_pscore<<<N, 64, 0, stream>>>(B_X, F(24), F(25), F(26), F(27), mask, nsc);
    k_gmax<<<(N + 255) / 256, 256, 0, stream>>>(nsc, batch, gmax, N);
    k_gsum<<<(N + 255) / 256, 256, 0, stream>>>(nsc, batch, gmax, gsum, N);
    k_graphlet<<<(int)(((long)N * 128 + 255) / 256), 256, 0, stream>>>(B_X, nsc, gsum, batch, glbuf, N);

    // --- heads ---
    k_heads<<<NGRAPH, 128, 0, stream>>>(
        glbuf, qemb,
        F(46), F(47), F(48), F(49),                    // qenc
        F(28), F(29), F(30), F(31), F(32), F(33),      // structural
        F(34), F(35), F(36), F(37), F(38), F(39),      // semantic
        F(40), F(41), F(42), F(43), F(44), F(45),      // logic
        (float*)d_out);

    (void)in_sizes; (void)n_in; (void)out_size; (void)ws_size;
}